// LorentzianMemoryAttention_73186242723947
// MI455X (gfx1250) — compile-verified
//
#include <hip/hip_runtime.h>
#include <hip/hip_bf16.h>

// Problem constants (match reference)
#define BQ   32
#define NMEM 4096
#define DM   1024
#define DL   64

#define TILE_ROWS 16   // rows of x_mem per TDM tile (64 KB)
#define TILES     16   // tiles per chunk
#define CHUNK     256  // rows of x_mem per workgroup

typedef float v2f __attribute__((ext_vector_type(2)));
typedef float v8f __attribute__((ext_vector_type(8)));

#if defined(__has_builtin)
#if __has_builtin(__builtin_amdgcn_tensor_load_to_lds)
#define HAVE_TDM 1
#endif
#endif
#ifndef HAVE_TDM
#define HAVE_TDM 0
#endif

#if HAVE_TDM
typedef unsigned int u32x4 __attribute__((ext_vector_type(4)));
typedef int          i32x4t __attribute__((ext_vector_type(4)));
typedef int          i32x8t __attribute__((ext_vector_type(8)));

// Issue one TDM load of a [TILE_ROWS x 1024] f32 tile (row-contiguous) to LDS.
// D# packing per CDNA5 ISA ch.8: group0 = {count/type, lds_addr, global_addr},
// group1 = {data_size, tensor_dim0/1, tile_dim0/1, tensor_dim0_stride}.
__device__ __forceinline__ void tdm_load_tile(const float* gptr, float* lptr) {
  unsigned long long ga = (unsigned long long)(size_t)gptr;
  unsigned lds_off = (unsigned)(size_t)lptr;         // low 32 bits = LDS offset
  u32x4 g0;
  g0[0] = 1u;                                        // count=1, user mode
  g0[1] = lds_off;                                   // lds_addr (bytes)
  g0[2] = (unsigned)(ga & 0xFFFFFFFFull);            // global_addr[31:0]
  g0[3] = (unsigned)((ga >> 32) & 0x1FFFFFFull)      // global_addr[56:32]
        | (2u << 30);                                // type=2 ("image")
  i32x8t g1;
  g1[0] = (2 << 16);                                 // data_size=2 -> 4 bytes
  g1[1] = (int)((unsigned)(DM & 0xFFFF) << 16);      // tensor_dim0[15:0]=1024
  g1[2] = (int)((unsigned)(TILE_ROWS & 0xFFFF) << 16); // tensor_dim1[15:0]
  g1[3] = (int)((unsigned)(DM & 0xFFFF) << 16);      // tile_dim0=1024
  g1[4] = TILE_ROWS;                                 // tile_dim1
  g1[5] = DM;                                        // tensor_dim0_stride=1024
  g1[6] = 0;
  g1[7] = 0;
  i32x4t z4 = {0, 0, 0, 0};
#if __clang_major__ >= 23
  i32x8t z8 = {0, 0, 0, 0, 0, 0, 0, 0};
  __builtin_amdgcn_tensor_load_to_lds(g0, g1, z4, z4, z8, 0);
#else
  __builtin_amdgcn_tensor_load_to_lds(g0, g1, z4, z4, 0);
#endif
}
#endif  // HAVE_TDM

// ---------------------------------------------------------------------------
// Generic f32 WMMA GEMM:  C[M x N] (+)= A[M x K] * B[K x N]
// Block = 128 threads = 4 waves; each wave one 16x16 tile via chained
// V_WMMA_F32_16X16X4_F32. grid = (N/64, M/16). TRANSB: B stored [N x K].
// ---------------------------------------------------------------------------
template <bool TRANSB>
__global__ __launch_bounds__(128)
void k_gemm_f32_wmma(const float* __restrict__ A, int lda,
                     const float* __restrict__ Bm, int ldb,
                     float* __restrict__ C, int ldc,
                     int Kdim, int accumulate) {
  const int lane = threadIdx.x & 31;
  const int wave = threadIdx.x >> 5;
  const int m0 = blockIdx.y * 16;
  const int n0 = (blockIdx.x * 4 + wave) * 16;
  const int row = lane & 15;
  const int hi  = lane >> 4;

  v8f c = {};
  if (accumulate) {
#pragma unroll
    for (int r = 0; r < 8; ++r)
      c[r] = C[(size_t)(m0 + r + hi * 8) * ldc + (n0 + row)];
  }

  for (int k = 0; k < Kdim; k += 4) {
    v2f a, b;
    const float* ap = A + (size_t)(m0 + row) * lda + k + hi * 2;
    a[0] = ap[0];
    a[1] = ap[1];
    if (TRANSB) {
      const float* bp = Bm + (size_t)(n0 + row) * ldb + k + hi * 2;
      b[0] = bp[0];
      b[1] = bp[1];
    } else {
      const float* bp = Bm + (size_t)(k + hi * 2) * ldb + (n0 + row);
      b[0] = bp[0];
      b[1] = bp[ldb];
    }
    c = __builtin_amdgcn_wmma_f32_16x16x4_f32(false, a, false, b,
                                              (short)0, c, false, false);
  }

#pragma unroll
  for (int r = 0; r < 8; ++r)
    C[(size_t)(m0 + r + hi * 8) * ldc + (n0 + row)] = c[r];
}

// ---------------------------------------------------------------------------
// q init: biases + t*Wt + t^2*Wtt + t*(z@Wzt^T); also builds zz[b] = z (x) z
// ---------------------------------------------------------------------------
__global__ __launch_bounds__(256)
void k_qinit(const float* __restrict__ z, const float* __restrict__ t,
             const float* __restrict__ bx, const float* __restrict__ bz,
             const float* __restrict__ bt, const float* __restrict__ bv,
             const float* __restrict__ Wt, const float* __restrict__ Wtt,
             const float* __restrict__ Wzt,
             float* __restrict__ q, float* __restrict__ zz) {
  const int b = blockIdx.x, tid = threadIdx.x;
  __shared__ float zl[DL];
  if (tid < DL) zl[tid] = z[b * DL + tid];
  __syncthreads();
  const float tb = t[b];
  for (int o = tid; o < DM; o += 256) {
    float acc = bx[o] + bz[o] + bt[o] + bv[o] + tb * Wt[o] + tb * tb * Wtt[o];
    float zt = 0.f;
#pragma unroll 8
    for (int j = 0; j < DL; ++j) zt += zl[j] * Wzt[o * DL + j];
    q[b * DM + o] = acc + tb * zt;
  }
  for (int idx = tid; idx < DL * DL; idx += 256)
    zz[b * DL * DL + idx] = zl[idx >> 6] * zl[idx & 63];
}

// ---------------------------------------------------------------------------
// Per-batch prep: qbk = q.bk ; norm1 = |z|^2 ; tau = 16*(1-norm1+eps)
// ---------------------------------------------------------------------------
__global__ __launch_bounds__(256)
void k_bprep(const float* __restrict__ q, const float* __restrict__ bk,
             const float* __restrict__ z,
             float* __restrict__ qbk, float* __restrict__ taub,
             float* __restrict__ norm1b) {
  const int b = blockIdx.x, tid = threadIdx.x;
  __shared__ float red[256];
  float a = 0.f;
  for (int d = tid; d < DM; d += 256) a += q[b * DM + d] * bk[d];
  red[tid] = a; __syncthreads();
  for (int st = 128; st > 0; st >>= 1) {
    if (tid < st) red[tid] += red[tid + st];
    __syncthreads();
  }
  if (tid == 0) qbk[b] = red[0];
  __syncthreads();
  float nz = 0.f;
  if (tid < DL) { float zv = z[b * DL + tid]; nz = zv * zv; }
  red[tid] = nz; __syncthreads();
  for (int st = 128; st > 0; st >>= 1) {
    if (tid < st) red[tid] += red[tid + st];
    __syncthreads();
  }
  if (tid == 0) {
    float n1 = red[0];
    norm1b[b] = n1;
    taub[b] = 16.0f * (1.0f - n1 + 1e-6f);   // sqrt(1024)/lam
  }
}

// ---------------------------------------------------------------------------
// Fused single pass over x_mem: scores + flash-style weighted accumulation.
// grid = (16 chunks, 32 b), block = 256 (8 waves, 2 rows/wave/tile).
// TDM double-buffers 16-row tiles into LDS; wave 0 drives the pipeline with
// s_wait_tensorcnt, block barriers publish tiles.
// ---------------------------------------------------------------------------
__global__ __launch_bounds__(256)
void k_fused(const float* __restrict__ x_mem, const float* __restrict__ z_mem,
             const float* __restrict__ t_mem, const float* __restrict__ z,
             const float* __restrict__ t, const float* __restrict__ qk,
             const float* __restrict__ qbk, const float* __restrict__ taub,
             const float* __restrict__ norm1b, const float* __restrict__ wscale_p,
             float* __restrict__ s, float* __restrict__ opart,
             float* __restrict__ mpart, float* __restrict__ lpart) {
  const int b = blockIdx.y, chunk = blockIdx.x;
  const int tid = threadIdx.x, lane = tid & 31, wave = tid >> 5;
  extern __shared__ float smem[];
  float* buf  = smem;                       // 2 * TILE_ROWS * DM
  float* osum = smem + 2 * TILE_ROWS * DM;  // DM
  float* red  = osum + DM;                  // 16

  // per-lane register slice of qk[b] (columns i*128 + lane*4 + c)
  float qreg[32];
#pragma unroll
  for (int i = 0; i < 8; ++i) {
    float4 qv = *(const float4*)(qk + b * DM + i * 128 + lane * 4);
    qreg[i * 4 + 0] = qv.x; qreg[i * 4 + 1] = qv.y;
    qreg[i * 4 + 2] = qv.z; qreg[i * 4 + 3] = qv.w;
  }
  const float z0 = z[b * DL + lane * 2], z1 = z[b * DL + lane * 2 + 1];
  const float wsc = wscale_p[0];
  const float tb = t[b], tau = taub[b], qb = qbk[b], n1 = norm1b[b];

  float m_w = -3.4e38f, l_w = 0.f;
  float acc[32];
#pragma unroll
  for (int k = 0; k < 32; ++k) acc[k] = 0.f;

  const float* xc = x_mem + ((size_t)b * NMEM + (size_t)chunk * CHUNK) * DM;

#if HAVE_TDM
  if (wave == 0) tdm_load_tile(xc, buf);
#endif

  for (int tl = 0; tl < TILES; ++tl) {
    float* cur;
#if HAVE_TDM
    cur = buf + (size_t)(tl & 1) * TILE_ROWS * DM;
    if (wave == 0) {
      if (tl + 1 < TILES) {
        tdm_load_tile(xc + (size_t)(tl + 1) * TILE_ROWS * DM,
                      buf + (size_t)((tl + 1) & 1) * TILE_ROWS * DM);
        __builtin_amdgcn_s_wait_tensorcnt(1);   // tile tl complete
      } else {
        __builtin_amdgcn_s_wait_tensorcnt(0);
      }
    }
    __syncthreads();                            // publish tile tl
#else
    cur = buf;
    __syncthreads();                            // protect previous compute
    {
      const float4* src = (const float4*)(xc + (size_t)tl * TILE_ROWS * DM);
      float4* dst = (float4*)cur;
      for (int idx = tid; idx < TILE_ROWS * (DM / 4); idx += 256) dst[idx] = src[idx];
    }
    __syncthreads();
#endif

#pragma unroll
    for (int rr = 0; rr < 2; ++rr) {
      const int rowin = wave * 2 + rr;
      const int n = chunk * CHUNK + tl * TILE_ROWS + rowin;
      const float* rp = cur + (size_t)rowin * DM;
      float va[32];
      float dot = 0.f;
#pragma unroll
      for (int i = 0; i < 8; ++i) {
        float4 v = *(const float4*)(rp + i * 128 + lane * 4);  // ds_load_b128
        va[i * 4 + 0] = v.x; va[i * 4 + 1] = v.y;
        va[i * 4 + 2] = v.z; va[i * 4 + 3] = v.w;
        dot += v.x * qreg[i * 4] + v.y * qreg[i * 4 + 1]
             + v.z * qreg[i * 4 + 2] + v.w * qreg[i * 4 + 3];
      }
      float2 zm = *(const float2*)(z_mem + ((size_t)b * NMEM + n) * DL + lane * 2);
      float dsq = (z0 - zm.x) * (z0 - zm.x) + (z1 - zm.y) * (z1 - zm.y);
      float n2  = zm.x * zm.x + zm.y * zm.y;
      for (int off = 16; off > 0; off >>= 1) {   // wave32 reduction
        dot += __shfl_down(dot, off, 32);
        dsq += __shfl_down(dsq, off, 32);
        n2  += __shfl_down(n2,  off, 32);
      }
      float sval = 0.f;
      if (lane == 0) {
        float denom = (1.0f - n1) * (1.0f - n2) + 1e-6f;
        float arg = fmaxf(1.0f + 2.0f * dsq / denom, 1.0f + 1e-6f);
        float dg = logf(arg + sqrtf(arg * arg - 1.0f));  // arccosh
        float wil = __expf(-wsc * dg);
        float dtv = tb - t_mem[b * NMEM + n];
        bool msk = (dtv > 0.0f) && (dg <= dtv);
        sval = msk ? (dot + qb) * wil / (tau + 1e-6f) : -1e9f;
        s[b * NMEM + n] = sval;
      }
      sval = __shfl(sval, 0, 32);                // broadcast (uniform per wave)
      if (sval > -1e8f) {                        // unmasked row
        if (sval > m_w) {                        // new running max: rescale
          float scl = __expf(m_w - sval);
          l_w = l_w * scl + 1.0f;
#pragma unroll
          for (int k = 0; k < 32; ++k) acc[k] = acc[k] * scl + va[k];
          m_w = sval;
        } else {
          float e = __expf(sval - m_w);
          l_w += e;
#pragma unroll
          for (int k = 0; k < 32; ++k) acc[k] += e * va[k];
        }
      }
    }
#if HAVE_TDM
    __syncthreads();   // all waves done with tile tl before it is overwritten
#endif
  }

  // ---- combine 8 waves within the block (LDS float atomics) ----
  if (lane == 0) red[wave] = m_w;
  __syncthreads();
  float m_blk = red[0];
#pragma unroll
  for (int wv = 1; wv < 8; ++wv) m_blk = fmaxf(m_blk, red[wv]);
  float f = __expf(m_w - m_blk);
  for (int j = tid; j < DM; j += 256) osum[j] = 0.f;
  if (lane == 0) red[8 + wave] = l_w * f;
  __syncthreads();
#pragma unroll
  for (int k = 0; k < 32; ++k) {
    int col = (k >> 2) * 128 + lane * 4 + (k & 3);
    atomicAdd(&osum[col], f * acc[k]);           // ds_add_f32
  }
  __syncthreads();
  float lblk = 0.f;
#pragma unroll
  for (int wv = 0; wv < 8; ++wv) lblk += red[8 + wv];
  for (int j = tid; j < DM; j += 256)
    opart[((size_t)(b * 16 + chunk)) * DM + j] = osum[j];
  if (tid == 0) { mpart[b * 16 + chunk] = m_blk; lpart[b * 16 + chunk] = lblk; }
}

// ---------------------------------------------------------------------------
// Softmax over N per batch -> weights (written straight to d_out)
// ---------------------------------------------------------------------------
__global__ __launch_bounds__(256)
void k_weights(const float* __restrict__ s, float* __restrict__ wout) {
  const int b = blockIdx.x, tid = threadIdx.x;
  __shared__ float red[256];
  float m = -3.4e38f;
  for (int n = tid; n < NMEM; n += 256) m = fmaxf(m, s[b * NMEM + n]);
  red[tid] = m; __syncthreads();
  for (int st = 128; st > 0; st >>= 1) {
    if (tid < st) red[tid] = fmaxf(red[tid], red[tid + st]);
    __syncthreads();
  }
  m = red[0]; __syncthreads();

  float ssum = 0.f;
  for (int n = tid; n < NMEM; n += 256) {
    float sv = s[b * NMEM + n];
    ssum += (sv > -1e8f) ? __expf(sv - m) : 0.0f;
  }
  red[tid] = ssum; __syncthreads();
  for (int st = 128; st > 0; st >>= 1) {
    if (tid < st) red[tid] += red[tid + st];
    __syncthreads();
  }
  const float S = red[0];
  const float dN = S + S * 1e-8f + 1e-30f;   // ref: /(sum(softmax*mask)+1e-8)
  for (int n = tid; n < NMEM; n += 256) {
    float sv = s[b * NMEM + n];
    float e = (sv > -1e8f) ? __expf(sv - m) : 0.0f;
    wout[b * NMEM + n] = e / dN;
  }
}

// ---------------------------------------------------------------------------
// Merge chunk partials -> h[b] (normalized), init out with sum(w)*bval.
// grid = (4, 32)
// ---------------------------------------------------------------------------
__global__ __launch_bounds__(256)
void k_combine(const float* __restrict__ opart, const float* __restrict__ mpart,
               const float* __restrict__ lpart, const float* __restrict__ bval,
               float* __restrict__ h, float* __restrict__ outp) {
  const int b = blockIdx.y;
  const int j = blockIdx.x * 256 + threadIdx.x;
  float m = -3.4e38f;
#pragma unroll
  for (int c = 0; c < 16; ++c) m = fmaxf(m, mpart[b * 16 + c]);
  float L = 0.f, acc = 0.f;
#pragma unroll
  for (int c = 0; c < 16; ++c) {
    float f = __expf(mpart[b * 16 + c] - m);
    L += f * lpart[b * 16 + c];
    acc += f * opart[((size_t)(b * 16 + c)) * DM + j];
  }
  const float dN = L + L * 1e-8f + 1e-30f;
  h[b * DM + j] = acc / dN;
  outp[b * DM + j] = (L / dN) * bval[j];   // sum(w) * bval
}

// ---------------------------------------------------------------------------
extern "C" void kernel_launch(void* const* d_in, const int* in_sizes, int n_in,
                              void* d_out, int out_size, void* d_ws, size_t ws_size,
                              hipStream_t stream) {
  const float* x      = (const float*)d_in[0];
  const float* z      = (const float*)d_in[1];
  const float* t      = (const float*)d_in[2];
  const float* x_mem  = (const float*)d_in[3];
  const float* z_mem  = (const float*)d_in[4];
  const float* t_mem  = (const float*)d_in[5];
  const float* v_feat = (const float*)d_in[6];
  const float* Wx     = (const float*)d_in[7];
  const float* bx     = (const float*)d_in[8];
  const float* Wz     = (const float*)d_in[9];
  const float* bz     = (const float*)d_in[10];
  const float* Wt     = (const float*)d_in[11];
  const float* bt     = (const float*)d_in[12];
  const float* Wv     = (const float*)d_in[13];
  const float* bv     = (const float*)d_in[14];
  const float* Wzz    = (const float*)d_in[15];
  const float* Wtt    = (const float*)d_in[16];
  const float* Wzt    = (const float*)d_in[17];
  const float* Wk     = (const float*)d_in[18];
  const float* bk     = (const float*)d_in[19];
  const float* Wval   = (const float*)d_in[20];
  const float* bval   = (const float*)d_in[21];
  const float* wscale = (const float*)d_in[22];

  float* outp = (float*)d_out;          // [32,1024]
  float* wout = outp + BQ * DM;         // [32,4096]

  float* ws = (float*)d_ws;
  float* q     = ws;                       // 32768
  float* zz    = q     + BQ * DM;          // 131072
  float* qk    = zz    + BQ * DL * DL;     // 32768
  float* s     = qk    + BQ * DM;          // 131072
  float* h     = s     + BQ * NMEM;        // 32768
  float* opart = h     + BQ * DM;          // 32*16*1024 = 524288
  float* mpart = opart + BQ * 16 * DM;     // 512
  float* lpart = mpart + BQ * 16;          // 512
  float* qbk   = lpart + BQ * 16;          // 32
  float* taub  = qbk   + BQ;               // 32
  float* n1b   = taub  + BQ;               // 32

  const dim3 gemmGrid(DM / 64, BQ / 16);   // N=1024, M=32

  // q = init(biases, t, Wzt) ; zz = z (x) z
  k_qinit<<<BQ, 256, 0, stream>>>(z, t, bx, bz, bt, bv, Wt, Wtt, Wzt, q, zz);

  // q += x@Wx + z@Wz + v_feat@Wv + zz@Wzz^T   (all WMMA)
  k_gemm_f32_wmma<false><<<gemmGrid, 128, 0, stream>>>(x, DM, Wx, DM, q, DM, DM, 1);
  k_gemm_f32_wmma<false><<<gemmGrid, 128, 0, stream>>>(z, DL, Wz, DM, q, DM, DL, 1);
  k_gemm_f32_wmma<false><<<gemmGrid, 128, 0, stream>>>(v_feat, DL, Wv, DM, q, DM, DL, 1);
  k_gemm_f32_wmma<true ><<<gemmGrid, 128, 0, stream>>>(zz, DL * DL, Wzz, DL * DL, q, DM, DL * DL, 1);

  // qk = q @ Wk^T  (WMMA, transB)
  k_gemm_f32_wmma<true ><<<gemmGrid, 128, 0, stream>>>(q, DM, Wk, DM, qk, DM, DM, 0);

  // per-batch scalars
  k_bprep<<<BQ, 256, 0, stream>>>(q, bk, z, qbk, taub, n1b);

  // fused single pass over x_mem: scores + flash accumulation (TDM staged)
  const size_t smemBytes = (size_t)(2 * TILE_ROWS * DM + DM + 16) * sizeof(float);
  k_fused<<<dim3(NMEM / CHUNK, BQ), 256, smemBytes, stream>>>(
      x_mem, z_mem, t_mem, z, t, qk, qbk, taub, n1b, wscale,
      s, opart, mpart, lpart);

  // softmax -> weights (direct to d_out)
  k_weights<<<BQ, 256, 0, stream>>>(s, wout);

  // merge chunk partials -> h; init out with sum(w)*bval
  k_combine<<<dim3(4, BQ), 256, 0, stream>>>(opart, mpart, lpart, bval, h, outp);

  // out += h @ Wval  (WMMA)
  k_gemm_f32_wmma<false><<<gemmGrid, 128, 0, stream>>>(h, DM, Wval, DM, outp, DM, DM, 1);
}